// GraphAttentionLayer_35416300323637
// MI455X (gfx1250) — compile-verified
//
#include <hip/hip_runtime.h>
#include <hip/hip_bf16.h>
#include <math.h>

typedef float v2f __attribute__((ext_vector_type(2)));
typedef float v8f __attribute__((ext_vector_type(8)));

#define GAT_B 16
#define GAT_N 1024
#define GAT_F 256
#define GAT_ALPHA 0.2f
#define GAT_NEGBIG -9000000000000000.0f

#define PSTR 1028              // P tile row stride (floats): banks (4m+k)%64, conflict-free
#define HSTR 260               // h tile row stride (floats): same trick
#define SSTR 264               // Wh slab row stride (floats): halves shifted by 16 banks
#define SLAB_K 16
#define SLAB_FLOATS (SLAB_K * SSTR)
#define SLAB_BYTES (SLAB_FLOATS * 4)

__device__ __forceinline__ void gat_wait_async0() {
#if __has_builtin(__builtin_amdgcn_s_wait_asynccnt)
  __builtin_amdgcn_s_wait_asynccnt(0);
#else
  asm volatile("s_wait_asynccnt 0x0" ::: "memory");
#endif
}

// Async-copy one 16x256 f32 k-slab of Wh_b into LDS (padded rows), 4 b128/lane.
__device__ __forceinline__ void gat_issue_slab(const float* __restrict__ Whb, int ks,
                                               uint32_t slab_lds_base, int t) {
#pragma unroll
  for (int q = 0; q < 4; ++q) {
    const int c   = q * 256 + t;        // 1024 16-byte chunks total
    const int row = c >> 6;             // 64 chunks per 256-float row
    const int col = (c & 63) * 4;
    const float* g = Whb + (size_t)(ks + row) * GAT_F + col;
    const uint32_t l = slab_lds_base + (uint32_t)(row * SSTR + col) * 4u;
    asm volatile("global_load_async_to_lds_b128 %0, %1, off"
                 :: "v"(l), "v"(g) : "memory");
  }
}

// ---------------------------------------------------------------------------
// Kernel 1: Wh = h @ W  (16384 x 256 x 256), f32 WMMA. 16 rows/WG, 8 waves.
// ---------------------------------------------------------------------------
__global__ __launch_bounds__(256) void gat_wh_gemm(const float* __restrict__ h,
                                                   const float* __restrict__ W,
                                                   float* __restrict__ Wh) {
  __shared__ float hs[16 * HSTR];
  const int rb = blockIdx.x * 16;
  const int t  = threadIdx.x;
#pragma unroll
  for (int r = 0; r < 16; ++r)
    hs[r * HSTR + t] = h[(size_t)(rb + r) * GAT_F + t];
  __syncthreads();

  const int wave = t >> 5;
  const int lane = t & 31;
  const int mn   = lane & 15;
  const int koff = (lane >> 4) * 2;
  const int nb0  = wave * 32;
  const int nb1  = nb0 + 16;
  v8f acc0 = {0,0,0,0,0,0,0,0};
  v8f acc1 = {0,0,0,0,0,0,0,0};

  for (int k = 0; k < GAT_F; k += 4) {
    v2f af, bf0, bf1;
    af.x  = hs[mn * HSTR + k + koff];
    af.y  = hs[mn * HSTR + k + koff + 1];
    bf0.x = W[(size_t)(k + koff)     * GAT_F + nb0 + mn];
    bf0.y = W[(size_t)(k + koff + 1) * GAT_F + nb0 + mn];
    bf1.x = W[(size_t)(k + koff)     * GAT_F + nb1 + mn];
    bf1.y = W[(size_t)(k + koff + 1) * GAT_F + nb1 + mn];
    acc0 = __builtin_amdgcn_wmma_f32_16x16x4_f32(false, af, false, bf0, (short)0, acc0, false, false);
    acc1 = __builtin_amdgcn_wmma_f32_16x16x4_f32(false, af, false, bf1, (short)0, acc1, false, false);
  }

  const int ms = (lane >> 4) * 8;
#pragma unroll
  for (int r = 0; r < 8; ++r) {
    Wh[(size_t)(rb + ms + r) * GAT_F + nb0 + mn] = acc0[r];
    Wh[(size_t)(rb + ms + r) * GAT_F + nb1 + mn] = acc1[r];
  }
}

// ---------------------------------------------------------------------------
// Kernel 2: f1/f2 rank-1 projections. One wave per row.
// ---------------------------------------------------------------------------
__global__ __launch_bounds__(256) void gat_fvec(const float* __restrict__ Wh,
                                                const float* __restrict__ a,
                                                float* __restrict__ f1,
                                                float* __restrict__ f2) {
  const int row  = blockIdx.x * 8 + (threadIdx.x >> 5);
  const int lane = threadIdx.x & 31;
  float s1 = 0.f, s2 = 0.f;
  for (int n = lane; n < GAT_F; n += 32) {
    const float w = Wh[(size_t)row * GAT_F + n];
    s1 += w * a[n];
    s2 += w * a[GAT_F + n];
  }
#pragma unroll
  for (int off = 16; off > 0; off >>= 1) {
    s1 += __shfl_xor(s1, off, 32);
    s2 += __shfl_xor(s2, off, 32);
  }
  if (lane == 0) { f1[row] = s1; f2[row] = s2; }
}

// ---------------------------------------------------------------------------
// Kernel 3: rank-1 scores -> softmax -> P @ Wh_b (WMMA, async-LDS double
// buffered) -> mask -> ELU.  One WG = (batch, 16-row tile), 8 waves.
// ---------------------------------------------------------------------------
__global__ __launch_bounds__(256) void gat_attn(const int*   __restrict__ adj,
                                                const float* __restrict__ Wh,
                                                const float* __restrict__ f1,
                                                const float* __restrict__ f2,
                                                const float* __restrict__ mask,
                                                float* __restrict__ out) {
  extern __shared__ float smem[];
  float* P_s  = smem;                       // [16][PSTR]
  float* slab = smem + 16 * PSTR;           // [2][SLAB_K][SSTR]
  __shared__ float rowsum_s[16];

  const int b    = blockIdx.y;
  const int ib   = blockIdx.x * 16;
  const int t    = threadIdx.x;
  const int wave = t >> 5;
  const int lane = t & 31;

  // ---- Phase A: scores + softmax weights (adj streamed once, b128) ----
  const float* f2b = f2 + b * GAT_N;
#pragma unroll
  for (int rr = 0; rr < 2; ++rr) {
    const int r = wave * 2 + rr;
    const int i = ib + r;
    const float f1r = f1[b * GAT_N + i];
    const int* adjrow = adj + ((size_t)b * GAT_N + i) * GAT_N;

    float mx = GAT_NEGBIG;
    for (int j0 = lane * 4; j0 < GAT_N; j0 += 128) {
      const int4   a4 = *(const int4*)(adjrow + j0);
      const float4 g4 = *(const float4*)(f2b + j0);
      float4 s4;
      float e;
      e = f1r + g4.x; e = (e > 0.f) ? e : GAT_ALPHA * e; s4.x = (a4.x > 0) ? e : GAT_NEGBIG;
      e = f1r + g4.y; e = (e > 0.f) ? e : GAT_ALPHA * e; s4.y = (a4.y > 0) ? e : GAT_NEGBIG;
      e = f1r + g4.z; e = (e > 0.f) ? e : GAT_ALPHA * e; s4.z = (a4.z > 0) ? e : GAT_NEGBIG;
      e = f1r + g4.w; e = (e > 0.f) ? e : GAT_ALPHA * e; s4.w = (a4.w > 0) ? e : GAT_NEGBIG;
      *(float4*)(P_s + r * PSTR + j0) = s4;
      mx = fmaxf(mx, fmaxf(fmaxf(s4.x, s4.y), fmaxf(s4.z, s4.w)));
    }
#pragma unroll
    for (int off = 16; off > 0; off >>= 1)
      mx = fmaxf(mx, __shfl_xor(mx, off, 32));

    float sum = 0.f;
    for (int j0 = lane * 4; j0 < GAT_N; j0 += 128) {
      float4 p4 = *(float4*)(P_s + r * PSTR + j0);
      p4.x = __expf(p4.x - mx);
      p4.y = __expf(p4.y - mx);
      p4.z = __expf(p4.z - mx);
      p4.w = __expf(p4.w - mx);
      *(float4*)(P_s + r * PSTR + j0) = p4;
      sum += p4.x + p4.y + p4.z + p4.w;
    }
#pragma unroll
    for (int off = 16; off > 0; off >>= 1)
      sum += __shfl_xor(sum, off, 32);
    if (lane == 0) rowsum_s[r] = sum;
  }

  // ---- Phase B: (16 x 1024) @ (1024 x 256), double-buffered async slabs ----
  const int mn   = lane & 15;
  const int koff = (lane >> 4) * 2;
  const int nb0  = wave * 32;
  const int nb1  = nb0 + 16;
  const float* Whb = Wh + (size_t)b * GAT_N * GAT_F;
  const uint32_t slab_base = (uint32_t)(uintptr_t)slab;
  v8f acc0 = {0,0,0,0,0,0,0,0};
  v8f acc1 = {0,0,0,0,0,0,0,0};

  gat_issue_slab(Whb, 0, slab_base, t);          // prefetch slab 0 -> buf 0
  int buf = 0;
  for (int s = 0; s < GAT_N / SLAB_K; ++s) {
    gat_wait_async0();                           // my slab-s loads landed
    __syncthreads();                             // everyone's landed
    if (s + 1 < GAT_N / SLAB_K)
      gat_issue_slab(Whb, (s + 1) * SLAB_K, slab_base + (uint32_t)((buf ^ 1) * SLAB_BYTES), t);

    const float* sb = slab + buf * SLAB_FLOATS;
    const int ks = s * SLAB_K;
#pragma unroll
    for (int kk = 0; kk < SLAB_K; kk += 4) {
      v2f af, bf0, bf1;
      af.x  = P_s[mn * PSTR + ks + kk + koff];
      af.y  = P_s[mn * PSTR + ks + kk + koff + 1];
      bf0.x = sb[(kk + koff)     * SSTR + nb0 + mn];
      bf0.y = sb[(kk + koff + 1) * SSTR + nb0 + mn];
      bf1.x = sb[(kk + koff)     * SSTR + nb1 + mn];
      bf1.y = sb[(kk + koff + 1) * SSTR + nb1 + mn];
      acc0 = __builtin_amdgcn_wmma_f32_16x16x4_f32(false, af, false, bf0, (short)0, acc0, false, false);
      acc1 = __builtin_amdgcn_wmma_f32_16x16x4_f32(false, af, false, bf1, (short)0, acc1, false, false);
    }
    __syncthreads();                             // all done reading buf
    buf ^= 1;
  }

  // ---- Epilogue: /rowsum, *mask, ELU ----
  const int ms = (lane >> 4) * 8;
#pragma unroll
  for (int r = 0; r < 8; ++r) {
    const int row = ms + r;
    const float sc = mask[b * GAT_N + ib + row] / rowsum_s[row];
    float v0 = acc0[r] * sc;
    v0 = (v0 > 0.f) ? v0 : expm1f(v0);
    out[((size_t)b * GAT_N + ib + row) * GAT_F + nb0 + mn] = v0;
    float v1 = acc1[r] * sc;
    v1 = (v1 > 0.f) ? v1 : expm1f(v1);
    out[((size_t)b * GAT_N + ib + row) * GAT_F + nb1 + mn] = v1;
  }
}

// ---------------------------------------------------------------------------
extern "C" void kernel_launch(void* const* d_in, const int* in_sizes, int n_in,
                              void* d_out, int out_size, void* d_ws, size_t ws_size,
                              hipStream_t stream) {
  const float* h    = (const float*)d_in[0];
  const int*   adj  = (const int*)  d_in[1];
  const float* mask = (const float*)d_in[2];
  const float* W    = (const float*)d_in[3];
  const float* a    = (const float*)d_in[4];
  float* out = (float*)d_out;

  float* Wh = (float*)d_ws;                                // 16 MB
  float* f1 = Wh + (size_t)GAT_B * GAT_N * GAT_F;           // 64 KB
  float* f2 = f1 + GAT_B * GAT_N;                           // 64 KB

  gat_wh_gemm<<<(GAT_B * GAT_N) / 16, 256, 0, stream>>>(h, W, Wh);
  gat_fvec  <<<(GAT_B * GAT_N) / 8, 256, 0, stream>>>(Wh, a, f1, f2);

  const size_t smem_bytes = (size_t)(16 * PSTR + 2 * SLAB_FLOATS) * sizeof(float);
  gat_attn<<<dim3(GAT_N / 16, GAT_B), 256, smem_bytes, stream>>>(
      adj, Wh, f1, f2, mask, out);
}